// SpectralSquaredFunctionWrapper_2783138808054
// MI455X (gfx1250) — compile-verified
//
#include <hip/hip_runtime.h>

typedef __attribute__((ext_vector_type(2))) float v2f;
typedef __attribute__((ext_vector_type(8))) float v8f;

namespace {
constexpr int Cc   = 64;          // channels (contraction dim)
constexpr int Kk   = 16;          // gram size
constexpr int Hh   = 128;
constexpr int Ww   = 128;
constexpr int HW   = Hh * Ww;     // 16384
constexpr int PIX  = 32;          // pixels per block == waves per block
constexpr int PAIRS = Cc * Kk;    // 1024 (c,k) pairs
constexpr int TS   = PIX + 1;     // LDS row stride (pad 32->33: bank-conflict-free)
constexpr float EPSV = 1e-6f;
}

__device__ __forceinline__ float bcast_lane(float v, int l) {
  // uniform-lane broadcast -> v_readlane (VALU pipe, SGPR operand for FMAs)
  return __builtin_bit_cast(float, __builtin_amdgcn_readlane(__builtin_bit_cast(int, v), l));
}
__device__ __forceinline__ float gather_lane(float v, int l) {
  // cross-lane gather -> ds_bpermute_b32 (LDS pipe; also used w/ uniform index
  // to offload broadcast traffic from the VALU issue slot)
  return __builtin_bit_cast(float, __builtin_amdgcn_ds_bpermute(l << 2, __builtin_bit_cast(int, v)));
}

extern "C" __global__ __launch_bounds__(PIX * 32, 1)
void gram_spectral_inverse(const float* __restrict__ x, float* __restrict__ out) {
  extern __shared__ float tile[];               // PAIRS * TS floats = 132 KB

  const int tid  = threadIdx.x;
  const int lane = tid & 31;
  const int wv   = tid >> 5;

  const int pb = blockIdx.x * PIX;              // first pixel of this block
  const int b  = pb >> 14;                      // pb / (H*W)
  const int h  = (pb >> 7) & 127;
  const int w0 = pb & 127;                      // multiple of 32

  const float* gbase = x + (size_t)b * (Cc * Kk * HW) + (size_t)h * Ww + w0;

  // ---- Stage [C*K pairs][32 pixels] slab. Each wave-instruction reads 32
  // consecutive floats (one 128B line) -> fully coalesced HBM streaming.
  #pragma unroll 8
  for (int it = 0; it < PAIRS / PIX; ++it) {
    const int pair = it * PIX + wv;
    tile[pair * TS + lane] = gbase[(size_t)pair * HW + lane];
  }
  __syncthreads();

  // ---- Per-wave Gram: sym = M * M^T (16x16, contracted over 64 channels).
  // f32 16x4 A-layout: lanes 0-15 hold M=lane K={c0,c0+1}; lanes 16-31 hold
  // K={c0+2,c0+3}. For A*A^T the B operand is the identical register pair.
  const int pix = wv;
  const int kx  = lane & 15;
  const int chi = (lane >> 4) << 1;             // 0 for low half-wave, 2 for high

  v8f acc = {};
  #pragma unroll
  for (int ch = 0; ch < 16; ++ch) {
    const int cA = ch * 4 + chi;
    v2f a;
    a.x = tile[(cA * 16 + kx) * TS + pix];      // k-stride 33 -> 32 distinct banks
    a.y = tile[((cA + 1) * 16 + kx) * TS + pix];
    acc = __builtin_amdgcn_wmma_f32_16x16x4_f32(
        /*neg_a=*/false, a, /*neg_b=*/false, a,
        /*c_mod=*/(short)0, acc, /*reuse_a=*/false, /*reuse_b=*/false);
  }

  // ---- Regularize: sym += EPS * I in C-fragment layout.
  // diag d<8  -> lane d,     VGPR d ; diag d>=8 -> lane d+16, VGPR d-8
  float s[8];
  #pragma unroll
  for (int r = 0; r < 8; ++r) {
    float v = acc[r];
    v += (lane == r || lane == r + 24) ? EPSV : 0.0f;
    s[r] = v;
  }

  // ---- Redistribute to column ownership for Gauss-Jordan:
  // lane L (<16) holds column L of sym (16 values); lane 16+j holds e_j.
  float aug[16];
  const bool left = lane < 16;
  #pragma unroll
  for (int i = 0; i < 8; ++i) {
    const float hi = gather_lane(s[i], kx + 16); // sym[i+8][kx] lives in lane kx+16
    aug[i]     = left ? s[i] : ((i == kx) ? 1.0f : 0.0f);
    aug[i + 8] = left ? hi   : ((i + 8 == kx) ? 1.0f : 0.0f);
  }

  // ---- Gauss-Jordan inverse of SPD 16x16 (no pivoting), fully unrolled.
  // [A | I] -> [I | A^-1]; lane 16+j ends with column j of A^-1.
  // Pivot reciprocal: v_rcp_f32 + one Newton step (saves the ~9-op IEEE div
  // chain). Column-k broadcasts alternate between v_readlane (VALU pipe) and
  // ds_bpermute (LDS pipe) to balance issue pressure.
  #pragma unroll
  for (int k = 0; k < 16; ++k) {
    const float p = bcast_lane(aug[k], k);      // pivot A[k][k]
    float invp = __builtin_amdgcn_rcpf(p);
    invp = fmaf(invp, fmaf(-p, invp, 1.0f), invp);  // NR refinement
    const float t = aug[k] * invp;              // scaled pivot row (per column)
    #pragma unroll
    for (int i = 0; i < 16; ++i) {
      if (i == k) continue;
      const float c = (i & 1) ? gather_lane(aug[i], k)   // LDS pipe
                              : bcast_lane(aug[i], k);   // VALU pipe
      aug[i] = fmaf(-c, t, aug[i]);
    }
    aug[k] = t;
  }

  // ---- Store. Inverse is symmetric, so column j == row j: lane 16+j writes
  // 16 consecutive floats (4x float4, 64B) -> the 16 lanes cover the whole
  // contiguous 1KB pixel output.
  if (!left) {
    const int j = kx;
    float* op = out + (size_t)(pb + pix) * 256 + j * 16;
    reinterpret_cast<float4*>(op)[0] = make_float4(aug[0],  aug[1],  aug[2],  aug[3]);
    reinterpret_cast<float4*>(op)[1] = make_float4(aug[4],  aug[5],  aug[6],  aug[7]);
    reinterpret_cast<float4*>(op)[2] = make_float4(aug[8],  aug[9],  aug[10], aug[11]);
    reinterpret_cast<float4*>(op)[3] = make_float4(aug[12], aug[13], aug[14], aug[15]);
  }
}

extern "C" void kernel_launch(void* const* d_in, const int* in_sizes, int n_in,
                              void* d_out, int out_size, void* d_ws, size_t ws_size,
                              hipStream_t stream) {
  (void)in_sizes; (void)n_in; (void)out_size; (void)d_ws; (void)ws_size;
  const float* x = (const float*)d_in[0];
  float* out = (float*)d_out;
  const int nblocks = (4 * HW) / PIX;                      // 65536 / 32 = 2048
  const size_t shmem = (size_t)PAIRS * TS * sizeof(float); // 135168 B
  hipLaunchKernelGGL(gram_spectral_inverse, dim3(nblocks), dim3(PIX * 32),
                     shmem, stream, x, out);
}